// POSExtractor_19155554140366
// MI455X (gfx1250) — compile-verified
//
#include <hip/hip_runtime.h>
#include <hip/hip_bf16.h>

// POS rPPG extractor, restructured to O(N):
//   stage1: 9 sliding 2nd-order stats per window via banded fp32 WMMA matmuls
//           -> per-window gains g_c[k] and offset t[k]
//   stage2: H[n] = sum_c x[n,c]*slidingsum(g_c)[n] - slidingsum(t)[n]
//           (same banded WMMA; zero-padded g-buffer handles edges)
// All WMMAs are V_WMMA_F32_16X16X4_F32 (exact fp32 math; the std-ratio is
// cancellation-sensitive, so low-precision WMMA flavors are not acceptable).
//
// A-matrix elements are produced BRANCHLESSLY: each lane's quantity selector
// is loop-invariant, so it is materialized as one-hot weight vectors outside
// the WMMA loop; the loop body is pure FMA + ds_load + wmma (EXEC untouched).

typedef __attribute__((ext_vector_type(2))) float v2f;
typedef __attribute__((ext_vector_type(8))) float v8f;

#define WIN 48

// Zero the 64-entry pads on both sides of the g-buffer so stage2 can index
// freely across the k in [0,K) boundary.
__global__ void pos_init_pad(float4* __restrict__ gbuf, int K) {
  int t = blockIdx.x * blockDim.x + threadIdx.x;
  float4 z; z.x = 0.f; z.y = 0.f; z.z = 0.f; z.w = 0.f;
  if (t < 64) gbuf[t] = z;
  else if (t < 128) gbuf[64 + K + (t - 64)] = z;
}

// ---- stage 1: per-window statistics -> g_c[k], t[k] ------------------------
// Block = 256 threads = 8 wave32s; block covers 128 windows, wave covers 16.
// Per wave: D[q,i] = sum_j A[q,j] * M[j,i], j over 64 samples (16 chunks of 4),
// M[j,i] = 1 iff i <= j < i+48  (banded sliding-window matrix).
// Quantities (A rows): 0:r 1:g 2:b 3:rr 4:rg 5:rb 6:gg 7:gb 8:bb, 9..15: 0.
__global__ __launch_bounds__(256) void pos_stage1(
    const float* __restrict__ x, const float* __restrict__ Wm,
    float4* __restrict__ gbuf, int N, int K) {
  __shared__ float sx[191 * 4];  // rows [kb, kb+190]: {r, g, b, 1}
  const int kb = blockIdx.x * 128;
  const int tid = threadIdx.x;
  for (int r = tid; r < 191; r += 256) {
    int n = kb + r;
    float vr = 0.f, vg = 0.f, vb = 0.f;
    if (n < N) { vr = x[3 * n + 0]; vg = x[3 * n + 1]; vb = x[3 * n + 2]; }
    sx[r * 4 + 0] = vr; sx[r * 4 + 1] = vg; sx[r * 4 + 2] = vb; sx[r * 4 + 3] = 1.f;
  }
  __syncthreads();

  const int lane = tid & 31;
  const int wave = tid >> 5;
  const int i = lane & 15;      // A row M (= quantity q) and D column N (= window)
  const int h = lane >> 4;      // lane half selects K pair
  const int kloc = wave * 16;   // this wave's first window, block-local

  // Per-lane factor indices packed 2 bits per q:  val(q) = ext[ia]*ext[ib]
  // with ext = {r, g, b, 1}; q>=9 -> zero via all-zero selector A.
  // ia: 0,1,2, 0,0,0, 1,1, 2, 3...   ib: 3,3,3, 0,1,2, 1,2, 2, 3...
  const int ia = (int)((0xFFFE5024u >> (2 * i)) & 3u);
  const int ib = (int)((0xFFFE993Fu >> (2 * i)) & 3u);
  const float zq = (i <= 8) ? 1.0f : 0.0f;   // kill padding quantities
  const float a0 = (ia == 0) ? zq : 0.f, a1 = (ia == 1) ? zq : 0.f;
  const float a2 = (ia == 2) ? zq : 0.f, a3 = (ia == 3) ? zq : 0.f;
  const float b0 = (ib == 0) ? 1.f : 0.f, b1 = (ib == 1) ? 1.f : 0.f;
  const float b2 = (ib == 2) ? 1.f : 0.f, b3 = (ib == 3) ? 1.f : 0.f;

  v8f acc = {};
#pragma unroll
  for (int t = 0; t < 16; ++t) {
    int j0 = 4 * t + 2 * h;     // K = 2h + reg per ISA 16x16x4 layout
    const float* r0 = &sx[(kloc + j0) * 4];
    const float* r1 = r0 + 4;
    float fa0 = a3 + a0 * r0[0] + a1 * r0[1] + a2 * r0[2];
    float fb0 = b3 + b0 * r0[0] + b1 * r0[1] + b2 * r0[2];
    float fa1 = a3 + a0 * r1[0] + a1 * r1[1] + a2 * r1[2];
    float fb1 = b3 + b0 * r1[0] + b1 * r1[1] + b2 * r1[2];
    v2f av, bv;
    av.x = fa0 * fb0;
    av.y = fa1 * fb1;
    bv.x = (j0     >= i && j0     < i + WIN) ? 1.0f : 0.0f;
    bv.y = (j0 + 1 >= i && j0 + 1 < i + WIN) ? 1.0f : 0.0f;
    acc = __builtin_amdgcn_wmma_f32_16x16x4_f32(false, av, false, bv,
                                                (short)0, acc, false, false);
  }

  // D rows 0..7 live in regs of lanes 0..15; row 8 (bb) is reg0 of lane+16.
  float Pbb = __shfl_xor(acc[0], 16, 32);

  if (h == 0) {
    int k = kb + kloc + i;
    if (k < K) {
      float Sr = acc[0], Sg = acc[1], Sb = acc[2];
      float Prr = acc[3], Prg = acc[4], Prb = acc[5], Pgg = acc[6], Pgb = acc[7];
      float mr = Sr * (1.0f / WIN), mg = Sg * (1.0f / WIN), mb = Sb * (1.0f / WIN);
      float w00 = Wm[0], w01 = Wm[1], w02 = Wm[2];
      float w10 = Wm[3], w11 = Wm[4], w12 = Wm[5];
      float ir = 1.0f / mr, ig = 1.0f / mg, ibb = 1.0f / mb;
      float a00 = w00 * ir, a01 = w01 * ig, a02 = w02 * ibb;
      float a10 = w10 * ir, a11 = w11 * ig, a12 = w12 * ibb;
      float Q0 = a00 * a00 * Prr + a01 * a01 * Pgg + a02 * a02 * Pbb
               + 2.0f * (a00 * a01 * Prg + a00 * a02 * Prb + a01 * a02 * Pgb);
      float Q1 = a10 * a10 * Prr + a11 * a11 * Pgg + a12 * a12 * Pbb
               + 2.0f * (a10 * a11 * Prg + a10 * a12 * Prb + a11 * a12 * Pgb);
      float s0 = w00 + w01 + w02, s1 = w10 + w11 + w12;
      float v0 = (Q0 - (float)WIN * s0 * s0) * (1.0f / (float)(WIN - 1));
      float v1 = (Q1 - (float)WIN * s1 * s1) * (1.0f / (float)(WIN - 1));
      float alpha = sqrtf(fmaxf(v0, 0.0f) / fmaxf(v1, 1e-30f));  // std0/std1
      float4 gout;
      gout.x = (w00 + alpha * w10) * ir;
      gout.y = (w01 + alpha * w11) * ig;
      gout.z = (w02 + alpha * w12) * ibb;
      gout.w = s0 + alpha * s1;     // constant term (0 for the POS matrix)
      gbuf[64 + k] = gout;
    }
  }
}

// ---- stage 2: overlap-add via the same banded WMMA -------------------------
// For output n: H[n] = sum_c x[n,c] * G_c[n] - T[n], with
// G/T[n] = sum of g/t over k in [n-47, n] (clamping handled by zero pads).
__global__ __launch_bounds__(256) void pos_stage2(
    const float* __restrict__ x, const float4* __restrict__ gbuf,
    float* __restrict__ H, int N) {
  __shared__ float4 sg[176];           // g entries k in [nb-47, nb+128]
  const int nb = blockIdx.x * 128;
  const int tid = threadIdx.x;
  for (int r = tid; r < 176; r += 256) {
    sg[r] = gbuf[64 + nb - 47 + r];    // lower pad suffices; last block may read past upper pad
  }
  __syncthreads();

  const int lane = tid & 31;
  const int wave = tid >> 5;
  const int i = lane & 15;
  const int h = lane >> 4;
  const int nloc = wave * 16;

  // One-hot component selector (rows 0..3 = g0,g1,g2,t; rows 4..15 = zero).
  const float c0 = (i == 0) ? 1.f : 0.f, c1 = (i == 1) ? 1.f : 0.f;
  const float c2 = (i == 2) ? 1.f : 0.f, c3 = (i == 3) ? 1.f : 0.f;

  v8f acc = {};
#pragma unroll
  for (int t = 0; t < 16; ++t) {
    int j0 = 4 * t + 2 * h;
    float4 v0 = sg[nloc + j0];
    float4 v1 = sg[nloc + j0 + 1];
    v2f av, bv;
    av.x = c0 * v0.x + c1 * v0.y + c2 * v0.z + c3 * v0.w;
    av.y = c0 * v1.x + c1 * v1.y + c2 * v1.z + c3 * v1.w;
    bv.x = (j0     >= i && j0     < i + WIN) ? 1.0f : 0.0f;
    bv.y = (j0 + 1 >= i && j0 + 1 < i + WIN) ? 1.0f : 0.0f;
    acc = __builtin_amdgcn_wmma_f32_16x16x4_f32(false, av, false, bv,
                                                (short)0, acc, false, false);
  }

  if (h == 0) {
    int n = nb + nloc + i;
    if (n < N) {
      float G0 = acc[0], G1 = acc[1], G2 = acc[2], T = acc[3];
      H[n] = x[3 * n + 0] * G0 + x[3 * n + 1] * G1 + x[3 * n + 2] * G2 - T;
    }
  }
}

extern "C" void kernel_launch(void* const* d_in, const int* in_sizes, int n_in,
                              void* d_out, int out_size, void* d_ws, size_t ws_size,
                              hipStream_t stream) {
  const float* x  = (const float*)d_in[0];   // rgbs [1, N, 3]
  const float* Wm = (const float*)d_in[1];   // W [2,3]
  // d_in[2] (bias b) cancels exactly under the final mean-centering.
  (void)n_in; (void)ws_size;

  const int N = in_sizes[0] / 3;
  const int K = N - WIN;
  float4* gbuf = (float4*)d_ws;              // [64 pad | K entries | 64 pad]
  float* H = (float*)d_out;
  (void)out_size;

  hipLaunchKernelGGL(pos_init_pad, dim3(1), dim3(128), 0, stream, gbuf, K);
  hipLaunchKernelGGL(pos_stage1, dim3((K + 127) / 128), dim3(256), 0, stream,
                     x, Wm, gbuf, N, K);
  hipLaunchKernelGGL(pos_stage2, dim3((N + 127) / 128), dim3(256), 0, stream,
                     x, gbuf, H, N);
}